// GroupedHybridRelationHead_29626684408280
// MI455X (gfx1250) — compile-verified
//
#include <hip/hip_runtime.h>
#include <hip/hip_bf16.h>
#include <math.h>
#include <stdint.h>

typedef float v2f __attribute__((ext_vector_type(2)));
typedef float v8f __attribute__((ext_vector_type(8)));

#define LN_EPS 1e-5f

// ---------------------------------------------------------------------------
// Async global->LDS 16-byte copy (GLOBAL_LOAD_ASYNC_TO_LDS_B128, ASYNCcnt-
// tracked). ROCm clang's builtin for this op has a non-portable descriptor
// signature, so issue the ISA instruction directly per CDNA5 ISA:
//   VDST  = VGPR holding LDS byte address (low 32 bits of flat shared ptr)
//   VADDR = VGPR pair holding 64-bit global address (GV mode, SADDR=off)
// ---------------------------------------------------------------------------
__device__ __forceinline__ void lds_async_b128(float* lds_dst, const float* gsrc) {
    uint32_t laddr = (uint32_t)(uintptr_t)lds_dst;  // LDS aperture: addr[31:0] = LDS offset
    uint64_t gaddr = (uint64_t)(uintptr_t)gsrc;
    asm volatile("global_load_async_to_lds_b128 %0, %1, off"
                 :
                 : "v"(laddr), "v"(gaddr)
                 : "memory");
}
__device__ __forceinline__ void lds_async_wait() {
    asm volatile("s_wait_asynccnt 0x0" ::: "memory");
}

__device__ __forceinline__ float gelu_exact(float x) {
    return 0.5f * x * (1.0f + erff(x * 0.70710678118654752f));
}
__device__ __forceinline__ float nan_to_num10(float x) {
    if (isnan(x)) return 0.0f;
    if (isinf(x)) return (x > 0.0f) ? 10.0f : -10.0f;
    return x;
}

// ---------------------------------------------------------------------------
// Kernel 1: per-row stats of z (sum, sumsq for first half zi and second half zj)
// grid = 512 (B*N), block = 256
// ---------------------------------------------------------------------------
__global__ __launch_bounds__(256) void stats_kernel(const float* __restrict__ z,
                                                    float* __restrict__ stats) {
    __shared__ float red[4][256];
    int row = blockIdx.x;
    int t = threadIdx.x;
    const float* zr = z + (size_t)row * 1024;
    float a = zr[t], b = zr[t + 256];          // zi half
    float c = zr[t + 512], d = zr[t + 768];    // zj half
    red[0][t] = a + b;
    red[1][t] = a * a + b * b;
    red[2][t] = c + d;
    red[3][t] = c * c + d * d;
    __syncthreads();
    for (int off = 128; off > 0; off >>= 1) {
        if (t < off) {
            red[0][t] += red[0][t + off];
            red[1][t] += red[1][t + off];
            red[2][t] += red[2][t + off];
            red[3][t] += red[3][t + off];
        }
        __syncthreads();
    }
    if (t == 0) {
        stats[row * 4 + 0] = red[0][0];
        stats[row * 4 + 1] = red[1][0];
        stats[row * 4 + 2] = red[2][0];
        stats[row * 4 + 3] = red[3][0];
    }
}

// ---------------------------------------------------------------------------
// Kernel 2: G[h] = sum_k g[k]*W_c[k,h];  C[h] = sum_k b[k]*W_c[k,h] + b_c[h]
// ---------------------------------------------------------------------------
__global__ __launch_bounds__(256) void gvec_kernel(const float* __restrict__ W_c,
                                                   const float* __restrict__ g,
                                                   const float* __restrict__ bvec,
                                                   const float* __restrict__ b_c,
                                                   float* __restrict__ G,
                                                   float* __restrict__ Cvec) {
    int h = threadIdx.x;
    float sg = 0.f, sb = 0.f;
    for (int k = 0; k < 1024; ++k) {
        float w = W_c[k * 256 + h];
        sg += g[k] * w;
        sb += bvec[k] * w;
    }
    G[h] = sg;
    Cvec[h] = sb + b_c[h];
}

// ---------------------------------------------------------------------------
// Kernel 3: repack U/V (c,H,R) class arrays into (H, 25*32) row-major matrices
// ---------------------------------------------------------------------------
__global__ __launch_bounds__(256) void repack_uv(const float* __restrict__ Ua,
                                                 const float* __restrict__ Us,
                                                 const float* __restrict__ Uc,
                                                 const float* __restrict__ Va,
                                                 const float* __restrict__ Vs,
                                                 const float* __restrict__ Vc,
                                                 float* __restrict__ Ucat,
                                                 float* __restrict__ Vcat) {
    int idx = blockIdx.x * blockDim.x + threadIdx.x;
    if (idx >= 256 * 800) return;
    int h = idx / 800, col = idx % 800;
    int cc = col / 32, r = col % 32;
    float u, v;
    if (cc < 3)      { u = Ua[cc * 8192 + h * 32 + r];       v = Va[cc * 8192 + h * 32 + r]; }
    else if (cc < 9) { u = Us[(cc - 3) * 8192 + h * 32 + r]; v = Vs[(cc - 3) * 8192 + h * 32 + r]; }
    else             { u = Uc[(cc - 9) * 8192 + h * 32 + r]; v = Vc[(cc - 9) * 8192 + h * 32 + r]; }
    Ucat[idx] = u;
    Vcat[idx] = v;
}

// ---------------------------------------------------------------------------
// Kernel 4: generic fp32 WMMA GEMM  C[16 rows/block, Ncols] = A @ diag(s) @ W
//   A tile staged via async global->LDS; per-k scale s folded into B fragments
//   (identity when colScale==nullptr). Uses V_WMMA_F32_16X16X4_F32.
// grid.x = rows/16, block = 256 (8 waves); wave w handles n-tiles w, w+8, ...
// ---------------------------------------------------------------------------
__global__ __launch_bounds__(256) void gemm_wmma_kernel(const float* __restrict__ A, int lda, int colOff,
                                                        const float* __restrict__ colScale,
                                                        const float* __restrict__ W, int ldw,
                                                        float* __restrict__ C, int ldc,
                                                        int K, int Ncols) {
    __shared__ float As[16 * 512];
    __shared__ float Ss[512];
    int row0 = blockIdx.x * 16;
    int tid = threadIdx.x;
    // stage 16 x K tile via ASYNCcnt path (16B chunks; all addresses 16B-aligned)
    int kchunks = K >> 2;
    for (int ch = tid; ch < 16 * kchunks; ch += blockDim.x) {
        int r = ch / kchunks, c4 = (ch % kchunks) << 2;
        lds_async_b128(&As[r * K + c4], A + (size_t)(row0 + r) * lda + colOff + c4);
    }
    for (int c = tid; c < K; c += blockDim.x)
        Ss[c] = colScale ? colScale[c] : 1.0f;
    lds_async_wait();
    __syncthreads();

    int wave = tid >> 5, lane = tid & 31;
    int m = lane & 15;
    int kOff = (lane < 16) ? 0 : 2;
    int mBase = (lane < 16) ? 0 : 8;
    int ntiles = Ncols >> 4;
    for (int nt = wave; nt < ntiles; nt += 8) {
        int n0 = nt << 4;
        v8f acc = {};
        for (int k = 0; k < K; k += 4) {
            v2f a, bf;
            a[0] = As[m * K + k + kOff];
            a[1] = As[m * K + k + kOff + 1];
            const float* wp = W + (size_t)(k + kOff) * ldw + n0 + m;
            bf[0] = wp[0] * Ss[k + kOff];
            bf[1] = wp[ldw] * Ss[k + kOff + 1];
            acc = __builtin_amdgcn_wmma_f32_16x16x4_f32(false, a, false, bf,
                                                        (short)0, acc, false, false);
        }
#pragma unroll
        for (int v = 0; v < 8; ++v) {
            int M = v + mBase;
            C[(size_t)(row0 + M) * ldc + n0 + m] = acc[v];
        }
    }
}

// ---------------------------------------------------------------------------
// Kernel 5: fused per-pair kernel. One block = 16 i-rows x 8 j-rows = 128 pairs.
// grid = (jt=16, it=8, b=4), block = 256 (8 waves). h lives entirely in LDS.
// ---------------------------------------------------------------------------
// LDS float offsets
#define SH_H    0          // 128*256
#define SH_G1   32768      // 128*64
#define SH_HI   40960      // 16*256
#define SH_AI   45056      // 16*256
#define SH_HJ   49152      // 8*256
#define SH_BJ   51200      // 8*256
#define SH_GV   53248      // 256
#define SH_CV   53504      // 256
#define SH_BG2  53760      // 256
#define SH_FG   54016      // 256
#define SH_FB   54272      // 256
#define SH_WG1  54528      // 8*64
#define SH_BG1  55040      // 64
#define SH_LGG  55104      // 8
#define SH_LGB  55112      // 8
#define SH_RSTD 55120      // 128
#define SH_MR   55248      // 128
#define SH_TOTAL 55376     // floats -> 221504 bytes

__global__ __launch_bounds__(256) void pair_kernel(
    const float* __restrict__ hi, const float* __restrict__ hj,
    const float* __restrict__ Ai, const float* __restrict__ Bj,
    const float* __restrict__ stats, const float* __restrict__ Gv,
    const float* __restrict__ Cvec, const float* __restrict__ boxes,
    const float* __restrict__ ln_g_g, const float* __restrict__ ln_g_b,
    const float* __restrict__ W_g1, const float* __restrict__ b_g1,
    const float* __restrict__ W_g2, const float* __restrict__ b_g2,
    const float* __restrict__ ln_f_g, const float* __restrict__ ln_f_b,
    const float* __restrict__ W_att, const float* __restrict__ b_att,
    const float* __restrict__ W_spa, const float* __restrict__ b_spa,
    const float* __restrict__ W_con, const float* __restrict__ b_con,
    const float* __restrict__ t1, const float* __restrict__ t2,
    const float* __restrict__ bb_att, const float* __restrict__ bb_spa,
    const float* __restrict__ bb_con, const float* __restrict__ alpha_att,
    const float* __restrict__ alpha_spa, const float* __restrict__ alpha_con,
    float* __restrict__ out) {
    extern __shared__ float smem[];
    int jt = blockIdx.x, it = blockIdx.y, bz = blockIdx.z;
    int i0 = it * 16, j0 = jt * 8;
    int tid = threadIdx.x;

    // ---- staging (big tiles via async global->LDS, 16B chunks) ----
    {
        const float* hiB = hi + (size_t)(bz * 128 + i0) * 256;
        const float* AiB = Ai + (size_t)(bz * 128 + i0) * 256;
        for (int ch = tid; ch < 16 * 64; ch += 256) {   // 16 rows x 64 chunks
            int off = ch << 2;
            lds_async_b128(&smem[SH_HI + off], hiB + off);
            lds_async_b128(&smem[SH_AI + off], AiB + off);
        }
        const float* hjB = hj + (size_t)(bz * 128 + j0) * 256;
        const float* BjB = Bj + (size_t)(bz * 128 + j0) * 256;
        for (int ch = tid; ch < 8 * 64; ch += 256) {    // 8 rows x 64 chunks
            int off = ch << 2;
            lds_async_b128(&smem[SH_HJ + off], hjB + off);
            lds_async_b128(&smem[SH_BJ + off], BjB + off);
        }
        smem[SH_GV + tid]  = Gv[tid];
        smem[SH_CV + tid]  = Cvec[tid];
        smem[SH_BG2 + tid] = b_g2[tid];
        smem[SH_FG + tid]  = ln_f_g[tid];
        smem[SH_FB + tid]  = ln_f_b[tid];
        for (int idx = tid; idx < 512; idx += 256) smem[SH_WG1 + idx] = W_g1[idx];
        if (tid < 64) smem[SH_BG1 + tid] = b_g1[tid];
        if (tid < 8) { smem[SH_LGG + tid] = ln_g_g[tid]; smem[SH_LGB + tid] = ln_g_b[tid]; }
        if (tid < 128) {
            int ri = bz * 128 + i0 + (tid >> 3);
            int rj = bz * 128 + j0 + (tid & 7);
            float s1 = stats[ri * 4 + 0] + stats[rj * 4 + 2];
            float s2 = stats[ri * 4 + 1] + stats[rj * 4 + 3];
            float m = s1 * (1.0f / 1024.0f);
            float var = s2 * (1.0f / 1024.0f) - m * m;
            float rs = rsqrtf(var + LN_EPS);
            smem[SH_RSTD + tid] = rs;
            smem[SH_MR + tid] = m * rs;
        }
        lds_async_wait();
    }
    __syncthreads();

    // ---- geometry branch: feats -> LN(8) -> 8x64 MLP -> gelu -> G1 ----
    if (tid < 128) {
        int p = tid;
        int i = i0 + (p >> 3), j = j0 + (p & 7);
        const float* bi = boxes + (size_t)(bz * 128 + i) * 4;
        const float* bj = boxes + (size_t)(bz * 128 + j) * 4;
        float x1i = bi[0], y1i = bi[1], x2i = bi[2], y2i = bi[3];
        float x1j = bj[0], y1j = bj[1], x2j = bj[2], y2j = bj[3];
        float wi = fmaxf(x2i - x1i, 1e-6f), hgi = fmaxf(y2i - y1i, 1e-6f);
        float wj = fmaxf(x2j - x1j, 1e-6f), hgj = fmaxf(y2j - y1j, 1e-6f);
        float cxi = (x1i + x2i) * 0.5f, cyi = (y1i + y2i) * 0.5f;
        float cxj = (x1j + x2j) * 0.5f, cyj = (y1j + y2j) * 0.5f;
        float dx = cxi - cxj, dy = cyi - cyj;
        float ldw = logf(wi / wj), ldh = logf(hgi / hgj);
        float xx1 = fmaxf(x1i, x1j), yy1 = fmaxf(y1i, y1j);
        float xx2 = fminf(x2i, x2j), yy2 = fminf(y2i, y2j);
        float ox = fmaxf(xx2 - xx1, 0.0f), oy = fmaxf(yy2 - yy1, 0.0f);
        float inter = ox * oy;
        float uni = fmaxf(wi * hgi + wj * hgj - inter, 1e-6f);
        float iou = inter / uni;
        float cd = sqrtf(dx * dx + dy * dy + 1e-12f);
        float f[8] = {dx, dy, ldw, ldh, iou, cd, ox, oy};
        float mn = 0.f;
#pragma unroll
        for (int k = 0; k < 8; ++k) { f[k] = nan_to_num10(f[k]); mn += f[k]; }
        mn *= 0.125f;
        float vr = 0.f;
#pragma unroll
        for (int k = 0; k < 8; ++k) { float d = f[k] - mn; vr += d * d; }
        vr *= 0.125f;
        float rs = rsqrtf(vr + LN_EPS);
        float l8[8];
#pragma unroll
        for (int k = 0; k < 8; ++k)
            l8[k] = (f[k] - mn) * rs * smem[SH_LGG + k] + smem[SH_LGB + k];
        for (int c = 0; c < 64; ++c) {
            float acc = smem[SH_BG1 + c];
#pragma unroll
            for (int k = 0; k < 8; ++k) acc += l8[k] * smem[SH_WG1 + k * 64 + c];
            smem[SH_G1 + p * 64 + c] = gelu_exact(acc);
        }
    }
    __syncthreads();

    // ---- WMMA: (128 pairs x 64) @ W_g2(64 x 256), fused assembly of h ----
    {
        int wave = tid >> 5, lane = tid & 31;
        int m = lane & 15;
        int kOff = (lane < 16) ? 0 : 2;
        int mBase = (lane < 16) ? 0 : 8;
        const float* G1w = &smem[SH_G1 + wave * 16 * 64];
        for (int nt = 0; nt < 16; ++nt) {
            int n0 = nt << 4;
            v8f acc = {};
#pragma unroll
            for (int k = 0; k < 64; k += 4) {
                v2f a, bf;
                a[0] = G1w[m * 64 + k + kOff];
                a[1] = G1w[m * 64 + k + kOff + 1];
                const float* wp = W_g2 + (k + kOff) * 256 + n0 + m;
                bf[0] = wp[0];
                bf[1] = wp[256];
                acc = __builtin_amdgcn_wmma_f32_16x16x4_f32(false, a, false, bf,
                                                            (short)0, acc, false, false);
            }
            int n = n0 + m;
#pragma unroll
            for (int v = 0; v < 8; ++v) {
                int M = v + mBase;
                int p = wave * 16 + M;
                int ii = p >> 3, jj = p & 7;
                float hc = (smem[SH_AI + ii * 256 + n] + smem[SH_BJ + jj * 256 + n]) * smem[SH_RSTD + p]
                           - smem[SH_MR + p] * smem[SH_GV + n] + smem[SH_CV + n];
                hc = gelu_exact(hc);
                float hv = smem[SH_HI + ii * 256 + n] + smem[SH_HJ + jj * 256 + n] + hc
                           + acc[v] + smem[SH_BG2 + n];
                smem[SH_H + p * 256 + n] = hv;
            }
        }
    }
    __syncthreads();

    // ---- final LayerNorm over H=256 (wave-parallel, shfl reduction) ----
    {
        int wave = tid >> 5, lane = tid & 31;
        for (int pi = 0; pi < 16; ++pi) {
            int p = wave * 16 + pi;
            float s = 0.f, s2 = 0.f, vals[8];
#pragma unroll
            for (int k = 0; k < 8; ++k) {
                float x = smem[SH_H + p * 256 + lane + 32 * k];
                vals[k] = x; s += x; s2 += x * x;
            }
            for (int off = 16; off > 0; off >>= 1) {
                s += __shfl_xor(s, off, 32);
                s2 += __shfl_xor(s2, off, 32);
            }
            float mean = s * (1.0f / 256.0f);
            float var = s2 * (1.0f / 256.0f) - mean * mean;
            float rs = rsqrtf(var + LN_EPS);
#pragma unroll
            for (int k = 0; k < 8; ++k) {
                int n = lane + 32 * k;
                smem[SH_H + p * 256 + n] = (vals[k] - mean) * rs * smem[SH_FG + n] + smem[SH_FB + n];
            }
        }
    }
    __syncthreads();

    // ---- outputs: h@W + b + sigmoid(alpha)*(t1.t2 + bb) for 25 classes ----
    {
        float sa = 1.0f / (1.0f + expf(-alpha_att[0]));
        float ss = 1.0f / (1.0f + expf(-alpha_spa[0]));
        float sc = 1.0f / (1.0f + expf(-alpha_con[0]));
        float* att = out;
        float* spa = out + 196608;
        float* con = out + 589824;
        for (int idx = tid; idx < 128 * 25; idx += 256) {
            int p = idx / 25, cc = idx % 25;
            int i = i0 + (p >> 3), j = j0 + (p & 7);
            const float* hr = &smem[SH_H + p * 256];
            const float* Wsel; int ldW, c2; float bias, bbv, sg; float* op;
            size_t pairIdx = ((size_t)(bz * 128 + i)) * 128 + j;
            if (cc < 3) {
                Wsel = W_att; ldW = 3; c2 = cc;
                bias = b_att[c2]; bbv = bb_att[c2]; sg = sa;
                op = att + pairIdx * 3 + c2;
            } else if (cc < 9) {
                Wsel = W_spa; ldW = 6; c2 = cc - 3;
                bias = b_spa[c2]; bbv = bb_spa[c2]; sg = ss;
                op = spa + pairIdx * 6 + c2;
            } else {
                Wsel = W_con; ldW = 16; c2 = cc - 9;
                bias = b_con[c2]; bbv = bb_con[c2]; sg = sc;
                op = con + pairIdx * 16 + c2;
            }
            float dot = 0.f;
            for (int n = 0; n < 256; ++n) dot += hr[n] * Wsel[n * ldW + c2];
            const float* t1r = t1 + ((size_t)(bz * 128 + i)) * 800 + cc * 32;
            const float* t2r = t2 + ((size_t)(bz * 128 + j)) * 800 + cc * 32;
            float d2 = 0.f;
#pragma unroll
            for (int r = 0; r < 32; ++r) d2 += t1r[r] * t2r[r];
            *op = dot + bias + sg * (d2 + bbv);
        }
    }
}

// ---------------------------------------------------------------------------
extern "C" void kernel_launch(void* const* d_in, const int* in_sizes, int n_in,
                              void* d_out, int out_size, void* d_ws, size_t ws_size,
                              hipStream_t stream) {
    const float* z       = (const float*)d_in[0];
    const float* boxes   = (const float*)d_in[1];
    const float* W_s     = (const float*)d_in[2];
    const float* W_o     = (const float*)d_in[3];
    const float* ln_c_g  = (const float*)d_in[4];
    const float* ln_c_b  = (const float*)d_in[5];
    const float* W_c     = (const float*)d_in[6];
    const float* b_c     = (const float*)d_in[7];
    const float* ln_g_g  = (const float*)d_in[8];
    const float* ln_g_b  = (const float*)d_in[9];
    const float* W_g1    = (const float*)d_in[10];
    const float* b_g1    = (const float*)d_in[11];
    const float* W_g2    = (const float*)d_in[12];
    const float* b_g2    = (const float*)d_in[13];
    const float* ln_f_g  = (const float*)d_in[14];
    const float* ln_f_b  = (const float*)d_in[15];
    const float* W_att   = (const float*)d_in[16];
    const float* b_att   = (const float*)d_in[17];
    const float* W_spa   = (const float*)d_in[18];
    const float* b_spa   = (const float*)d_in[19];
    const float* W_con   = (const float*)d_in[20];
    const float* b_con   = (const float*)d_in[21];
    const float* U_att   = (const float*)d_in[22];
    const float* V_att   = (const float*)d_in[23];
    const float* bb_att  = (const float*)d_in[24];
    const float* U_spa   = (const float*)d_in[25];
    const float* V_spa   = (const float*)d_in[26];
    const float* bb_spa  = (const float*)d_in[27];
    const float* U_con   = (const float*)d_in[28];
    const float* V_con   = (const float*)d_in[29];
    const float* bb_con  = (const float*)d_in[30];
    const float* a_att   = (const float*)d_in[31];
    const float* a_spa   = (const float*)d_in[32];
    const float* a_con   = (const float*)d_in[33];

    float* ws    = (float*)d_ws;
    float* stats = ws;                  // 2048
    float* hi    = stats + 2048;        // 512*256
    float* hj    = hi + 131072;
    float* Ai    = hj + 131072;
    float* Bj    = Ai + 131072;
    float* t1    = Bj + 131072;         // 512*800
    float* t2    = t1 + 409600;
    float* Ucat  = t2 + 409600;         // 256*800
    float* Vcat  = Ucat + 204800;
    float* Gv    = Vcat + 204800;       // 256
    float* Cvec  = Gv + 256;            // 256

    stats_kernel<<<512, 256, 0, stream>>>(z, stats);
    gvec_kernel<<<1, 256, 0, stream>>>(W_c, ln_c_g, ln_c_b, b_c, Gv, Cvec);
    repack_uv<<<(256 * 800 + 255) / 256, 256, 0, stream>>>(U_att, U_spa, U_con,
                                                           V_att, V_spa, V_con, Ucat, Vcat);

    // hi = zi @ W_s ; hj = zj @ W_o ; Ai = zi @ diag(g_top) W_c_top ; Bj = zj @ diag(g_bot) W_c_bot
    gemm_wmma_kernel<<<32, 256, 0, stream>>>(z, 1024, 0,   nullptr,      W_s, 256, hi, 256, 512, 256);
    gemm_wmma_kernel<<<32, 256, 0, stream>>>(z, 1024, 512, nullptr,      W_o, 256, hj, 256, 512, 256);
    gemm_wmma_kernel<<<32, 256, 0, stream>>>(z, 1024, 0,   ln_c_g,       W_c, 256, Ai, 256, 512, 256);
    gemm_wmma_kernel<<<32, 256, 0, stream>>>(z, 1024, 512, ln_c_g + 512, W_c + 512 * 256, 256, Bj, 256, 512, 256);
    // t1 = hi @ Ucat ; t2 = hj @ Vcat
    gemm_wmma_kernel<<<32, 256, 0, stream>>>(hi, 256, 0, nullptr, Ucat, 800, t1, 800, 256, 800);
    gemm_wmma_kernel<<<32, 256, 0, stream>>>(hj, 256, 0, nullptr, Vcat, 800, t2, 800, 256, 800);

    size_t smemBytes = (size_t)SH_TOTAL * sizeof(float);
    (void)hipFuncSetAttribute((const void*)pair_kernel,
                              hipFuncAttributeMaxDynamicSharedMemorySize, (int)smemBytes);
    pair_kernel<<<dim3(16, 8, 4), 256, smemBytes, stream>>>(
        hi, hj, Ai, Bj, stats, Gv, Cvec, boxes, ln_g_g, ln_g_b,
        W_g1, b_g1, W_g2, b_g2, ln_f_g, ln_f_b,
        W_att, b_att, W_spa, b_spa, W_con, b_con,
        t1, t2, bb_att, bb_spa, bb_con, a_att, a_spa, a_con,
        (float*)d_out);
}